// QCNNi_79680233275630
// MI455X (gfx1250) — compile-verified
//
#include <hip/hip_runtime.h>

typedef __attribute__((ext_vector_type(2))) float v2f;
typedef __attribute__((ext_vector_type(8))) float v8f;

// D = A(16x4) * B(4x16) + C, fp32 WMMA (wave32)
__device__ __forceinline__ v8f wmma_f32(v2f a, v2f b, v8f c) {
  return __builtin_amdgcn_wmma_f32_16x16x4_f32(false, a, false, b, (short)0, c,
                                               false, false);
}

__device__ __forceinline__ void cmul(float ar, float ai, float br, float bi,
                                     float& cr, float& ci) {
  cr = ar * br - ai * bi;
  ci = ar * bi + ai * br;
}

// 2x2 complex matmul C = A*B (row-major, idx = r*2+c)
__device__ __forceinline__ void mm2(const float* Ar, const float* Ai,
                                    const float* Br, const float* Bi,
                                    float* Cr, float* Ci) {
#pragma unroll
  for (int r = 0; r < 2; ++r)
#pragma unroll
    for (int c = 0; c < 2; ++c) {
      float sr = 0.f, si = 0.f;
#pragma unroll
      for (int k = 0; k < 2; ++k) {
        float pr, pi;
        cmul(Ar[r * 2 + k], Ai[r * 2 + k], Br[k * 2 + c], Bi[k * 2 + c], pr, pi);
        sr += pr;
        si += pi;
      }
      Cr[r * 2 + c] = sr;
      Ci[r * 2 + c] = si;
    }
}

__device__ __forceinline__ void rotx(float t, float* Re, float* Im) {
  float c = cosf(0.5f * t), s = sinf(0.5f * t);
  Re[0] = c; Re[1] = 0.f; Re[2] = 0.f; Re[3] = c;
  Im[0] = 0.f; Im[1] = -s; Im[2] = -s; Im[3] = 0.f;
}
__device__ __forceinline__ void roty(float t, float* Re, float* Im) {
  float c = cosf(0.5f * t), s = sinf(0.5f * t);
  Re[0] = c; Re[1] = -s; Re[2] = s; Re[3] = c;
  Im[0] = 0.f; Im[1] = 0.f; Im[2] = 0.f; Im[3] = 0.f;
}

// ---- fused 4-qubit (16x16 complex) gate applications ------------------------
// Group A: qubits 0-3  = flat bits 11..8 : S' = G @ S,  S viewed (16 x 256)
__device__ __forceinline__ void apply_groupA(const float* GR, const float* GI,
                                             float* Sr, float* Si, int wave,
                                             int lo, int hi) {
#pragma unroll
  for (int t = 0; t < 2; ++t) {
    const int c0 = (wave * 2 + t) * 16;
    v8f aR = {}; v8f aI = {};
#pragma unroll
    for (int kb = 0; kb < 16; kb += 4) {
      const int k0 = kb + 2 * hi;
      v2f gr, gi, gn, br, bi;
      gr[0] = GR[lo * 16 + k0];      gr[1] = GR[lo * 16 + k0 + 1];
      gi[0] = GI[lo * 16 + k0];      gi[1] = GI[lo * 16 + k0 + 1];
      gn[0] = -gi[0];                gn[1] = -gi[1];
      br[0] = Sr[k0 * 256 + c0 + lo];       br[1] = Sr[(k0 + 1) * 256 + c0 + lo];
      bi[0] = Si[k0 * 256 + c0 + lo];       bi[1] = Si[(k0 + 1) * 256 + c0 + lo];
      aR = wmma_f32(gr, br, aR);
      aR = wmma_f32(gn, bi, aR);
      aI = wmma_f32(gr, bi, aI);
      aI = wmma_f32(gi, br, aI);
    }
#pragma unroll
    for (int v = 0; v < 8; ++v) {
      const int m = v + 8 * hi;
      Sr[m * 256 + c0 + lo] = aR[v];
      Si[m * 256 + c0 + lo] = aI[v];
    }
  }
}

// Group B: qubits 4-7 = flat bits 7..4 : per h slice, S'_h = G @ S_h (16x16)
__device__ __forceinline__ void apply_groupB(const float* GR, const float* GI,
                                             float* Sr, float* Si, int wave,
                                             int lo, int hi) {
#pragma unroll
  for (int t = 0; t < 2; ++t) {
    const int base = (wave * 2 + t) * 256;
    v8f aR = {}; v8f aI = {};
#pragma unroll
    for (int kb = 0; kb < 16; kb += 4) {
      const int k0 = kb + 2 * hi;
      v2f gr, gi, gn, br, bi;
      gr[0] = GR[lo * 16 + k0];      gr[1] = GR[lo * 16 + k0 + 1];
      gi[0] = GI[lo * 16 + k0];      gi[1] = GI[lo * 16 + k0 + 1];
      gn[0] = -gi[0];                gn[1] = -gi[1];
      br[0] = Sr[base + k0 * 16 + lo];      br[1] = Sr[base + (k0 + 1) * 16 + lo];
      bi[0] = Si[base + k0 * 16 + lo];      bi[1] = Si[base + (k0 + 1) * 16 + lo];
      aR = wmma_f32(gr, br, aR);
      aR = wmma_f32(gn, bi, aR);
      aI = wmma_f32(gr, bi, aI);
      aI = wmma_f32(gi, br, aI);
    }
#pragma unroll
    for (int v = 0; v < 8; ++v) {
      const int m = v + 8 * hi;
      Sr[base + m * 16 + lo] = aR[v];
      Si[base + m * 16 + lo] = aI[v];
    }
  }
}

// Group C: qubits 8-11 = flat bits 3..0 : S' = S @ G^T, S viewed (256 x 16)
__device__ __forceinline__ void apply_groupC(const float* GR, const float* GI,
                                             float* Sr, float* Si, int wave,
                                             int lo, int hi) {
#pragma unroll
  for (int t = 0; t < 2; ++t) {
    const int r0 = (wave * 2 + t) * 16;
    v8f aR = {}; v8f aI = {};
#pragma unroll
    for (int kb = 0; kb < 16; kb += 4) {
      const int k0 = kb + 2 * hi;
      v2f ar, ai, gtr, gti, gtn;
      ar[0] = Sr[(r0 + lo) * 16 + k0];   ar[1] = Sr[(r0 + lo) * 16 + k0 + 1];
      ai[0] = Si[(r0 + lo) * 16 + k0];   ai[1] = Si[(r0 + lo) * 16 + k0 + 1];
      // B(k,n) = G[n][k] -> same indexing pattern as an A-load from G
      gtr[0] = GR[lo * 16 + k0];         gtr[1] = GR[lo * 16 + k0 + 1];
      gti[0] = GI[lo * 16 + k0];         gti[1] = GI[lo * 16 + k0 + 1];
      gtn[0] = -gti[0];                  gtn[1] = -gti[1];
      aR = wmma_f32(ar, gtr, aR);
      aR = wmma_f32(ai, gtn, aR);
      aI = wmma_f32(ar, gti, aI);
      aI = wmma_f32(ai, gtr, aI);
    }
#pragma unroll
    for (int v = 0; v < 8; ++v) {
      const int m = v + 8 * hi;
      Sr[(r0 + m) * 16 + lo] = aR[v];
      Si[(r0 + m) * 16 + lo] = aI[v];
    }
  }
}

// ISWAP on flat bits (b1, b1+1): swap amps at base|1<<b1 <-> base|2<<b1, *i
__device__ __forceinline__ void iswap_gate(float* Sr, float* Si, int b1,
                                           int tid) {
#pragma unroll
  for (int s = 0; s < 4; ++s) {
    int j = tid + 256 * s;  // 0..1023
    int low = j & ((1 << b1) - 1);
    int base = ((j >> b1) << (b1 + 2)) | low;
    int xp = base | (1 << b1);
    int yp = base | (2 << b1);
    float xr = Sr[xp], xi = Si[xp];
    float yr = Sr[yp], yi = Si[yp];
    Sr[xp] = -yi; Si[xp] = yr;   // i * old[y]
    Sr[yp] = -xi; Si[yp] = xr;   // i * old[x]
  }
}

__global__ __launch_bounds__(256) void qcnn_kernel(
    const float* __restrict__ x, const float* __restrict__ cw,
    const float* __restrict__ cb, const float* __restrict__ qw,
    float* __restrict__ out) {
  __shared__ float Sr[4096], Si[4096];         // state vector (SoA), 32 KB
  __shared__ float Gr[1536], Gi[1536];         // 2 blocks x 3 groups x 16x16
  __shared__ float ang[36];
  __shared__ float uRe[96], uIm[96];           // 24 composite 2x2 unitaries
  __shared__ float fRe[4], fIm[4];             // final qubit-0 gate

  const int tid = threadIdx.x;
  const int b = blockIdx.x;
  const int lane = tid & 31;
  const int wave = tid >> 5;
  const int lo = lane & 15;
  const int hi = lane >> 4;

  // ---- phase 0: zero state + conv -> angles --------------------------------
#pragma unroll
  for (int s = 0; s < 16; ++s) {
    Sr[tid + 256 * s] = 0.f;
    Si[tid + 256 * s] = 0.f;
  }
  if (tid < 36) {
    int i = tid / 6, j = tid % 6;
    float acc = cb[0];
#pragma unroll
    for (int p = 0; p < 4; ++p)
#pragma unroll
      for (int q = 0; q < 4; ++q)
        acc = fmaf(x[b * 196 + (2 * i + p) * 14 + (2 * j + q)], cw[p * 4 + q],
                   acc);
    ang[tid] = acc;
  }
  __syncthreads();

  // ---- phase 1: per-qubit composite 2x2 unitaries --------------------------
  if (tid < 24) {
    int blk = tid / 12, q = tid % 12;
    int i = q >> 1, off = (q & 1) * 3;
    float a0 = ang[i * 6 + off], a1 = ang[i * 6 + off + 1],
          a2 = ang[i * 6 + off + 2];
    float Ar[4], Ai[4], Br[4], Bi[4], Cr0[4], Ci0[4], Tr[4], Ti[4], Mr[4],
        Mi[4];
    if (blk == 0) {  // rx(a2)*ry(a1)*rx(a0)
      rotx(a0, Cr0, Ci0); roty(a1, Br, Bi); rotx(a2, Ar, Ai);
    } else {         // ry(a2)*rx(a1)*ry(a0)
      roty(a0, Cr0, Ci0); rotx(a1, Br, Bi); roty(a2, Ar, Ai);
    }
    mm2(Br, Bi, Cr0, Ci0, Tr, Ti);
    mm2(Ar, Ai, Tr, Ti, Mr, Mi);
#pragma unroll
    for (int e = 0; e < 4; ++e) {
      uRe[blk * 48 + q * 4 + e] = Mr[e];
      uIm[blk * 48 + q * 4 + e] = Mi[e];
    }
  } else if (tid == 24) {  // final gate: rx(w2)*ry(w1)*rx(w0)
    float Ar[4], Ai[4], Br[4], Bi[4], Cr0[4], Ci0[4], Tr[4], Ti[4], Mr[4],
        Mi[4];
    rotx(qw[0], Cr0, Ci0); roty(qw[1], Br, Bi); rotx(qw[2], Ar, Ai);
    mm2(Br, Bi, Cr0, Ci0, Tr, Ti);
    mm2(Ar, Ai, Tr, Ti, Mr, Mi);
#pragma unroll
    for (int e = 0; e < 4; ++e) { fRe[e] = Mr[e]; fIm[e] = Mi[e]; }
  } else if (tid == 31) {
    Sr[0] = 1.0f;  // |0...0>
  }
  __syncthreads();

  // ---- phase 2: kron 4 qubits -> 16x16 group unitaries (both blocks) -------
#pragma unroll
  for (int s = 0; s < 6; ++s) {
    int e = tid + 256 * s;              // 0..1535
    int blk = e / 768, rem = e % 768;
    int g = rem >> 8, idx = rem & 255;
    int m = idx >> 4, k = idx & 15;
    float pr = 1.f, pi = 0.f;
#pragma unroll
    for (int q = 0; q < 4; ++q) {
      int Q = 4 * g + q;
      int mb = (m >> (3 - q)) & 1, kb2 = (k >> (3 - q)) & 1;
      float ur = uRe[blk * 48 + Q * 4 + mb * 2 + kb2];
      float ui = uIm[blk * 48 + Q * 4 + mb * 2 + kb2];
      float nr, ni;
      cmul(pr, pi, ur, ui, nr, ni);
      pr = nr; pi = ni;
    }
    Gr[e] = pr; Gi[e] = pi;
  }
  __syncthreads();

  // ---- circuit: [gates, iswap chain] x 2 ------------------------------------
  for (int blk = 0; blk < 2; ++blk) {
    const float* GRb = Gr + blk * 768;
    const float* GIb = Gi + blk * 768;
    apply_groupA(GRb, GIb, Sr, Si, wave, lo, hi);
    __syncthreads();
    apply_groupB(GRb + 256, GIb + 256, Sr, Si, wave, lo, hi);
    __syncthreads();
    apply_groupC(GRb + 512, GIb + 512, Sr, Si, wave, lo, hi);
    __syncthreads();
    for (int i = 0; i < 11; ++i) {
      int b1 = (blk == 0) ? i : (10 - i);
      iswap_gate(Sr, Si, b1, tid);
      __syncthreads();
    }
  }

  // ---- final qubit-0 gate fused with |amp|^2 reduction ----------------------
  float f00r = fRe[0], f00i = fIm[0], f01r = fRe[1], f01i = fIm[1];
  float f10r = fRe[2], f10i = fIm[2], f11r = fRe[3], f11i = fIm[3];
  float p0 = 0.f, p1 = 0.f;
#pragma unroll
  for (int s = 0; s < 8; ++s) {
    int p = tid + 256 * s;  // pair (p, p+2048) differs in flat bit 11 (qubit 0)
    float a0r = Sr[p], a0i = Si[p];
    float a1r = Sr[p + 2048], a1i = Si[p + 2048];
    float n0r = f00r * a0r - f00i * a0i + f01r * a1r - f01i * a1i;
    float n0i = f00r * a0i + f00i * a0r + f01r * a1i + f01i * a1r;
    float n1r = f10r * a0r - f10i * a0i + f11r * a1r - f11i * a1i;
    float n1i = f10r * a0i + f10i * a0r + f11r * a1i + f11i * a1r;
    p0 += n0r * n0r + n0i * n0i;
    p1 += n1r * n1r + n1i * n1i;
  }
  __syncthreads();
  Gr[tid] = p0;  // reuse gate-matrix LDS as reduction scratch
  Gi[tid] = p1;
  __syncthreads();
  for (int off2 = 128; off2 > 0; off2 >>= 1) {
    if (tid < off2) {
      Gr[tid] += Gr[tid + off2];
      Gi[tid] += Gi[tid + off2];
    }
    __syncthreads();
  }
  if (tid == 0) {
    out[b * 2 + 0] = Gr[0];
    out[b * 2 + 1] = Gi[0];
  }
}

extern "C" void kernel_launch(void* const* d_in, const int* in_sizes, int n_in,
                              void* d_out, int out_size, void* d_ws,
                              size_t ws_size, hipStream_t stream) {
  const float* x = (const float*)d_in[0];    // (B,1,14,14)
  const float* cw = (const float*)d_in[1];   // (1,1,4,4)
  const float* cb = (const float*)d_in[2];   // (1,)
  const float* qw = (const float*)d_in[3];   // (3,)
  float* out = (float*)d_out;                // (B,2)
  int nb = in_sizes[0] / 196;
  qcnn_kernel<<<dim3(nb), dim3(256), 0, stream>>>(x, cw, cb, qw, out);
}